// LinearAttention3D_51599737094924
// MI455X (gfx1250) — compile-verified
//
#include <hip/hip_runtime.h>
#include <cstddef>

// ---------------------------------------------------------------------------
// LinearAttention3D for MI455X (gfx1250, wave32, WMMA).
// fp32 end-to-end using V_WMMA_F32_16X16X4_F32 (memory-bound problem; keep
// reference precision, still use the matrix pipe).
// ---------------------------------------------------------------------------

namespace {

constexpr int kB   = 2;
constexpr int kC   = 256;
constexpr int kH   = 8;
constexpr int kD   = 32;
constexpr int kG   = 8;
constexpr int kN   = 32768;            // 32^3 tokens
constexpr int kCPG = kC / kG;          // 32 channels per group
constexpr int kNT  = 64;               // tokens per workgroup tile
constexpr int kTilesPerBatch = kN / kNT;   // 512
constexpr int kLDXN = kNT + 4;         // LDS row stride for xn tile (pad)
constexpr int kLDT  = 18;              // LDS row stride for 16-token tiles (pad, even)

// Workspace layout (floats). Tiny (66 KB) -> always fits d_ws.
constexpr int WS_STAT  = 0;                          // [B*G][2] sum,sumsq
constexpr int WS_MRS   = 32;                         // [B*G][2] mean,rstd
constexpr int WS_CTX   = 64;                         // [B][H][D][D] = 16384
constexpr int WS_KSUM  = WS_CTX + kB * kH * kD * kD; // [B][C]     = 512
constexpr int WS_TOTAL = WS_KSUM + kB * kC;          // 16960 floats

typedef __attribute__((ext_vector_type(2))) float v2f;
typedef __attribute__((ext_vector_type(8))) float v8f;

__device__ __forceinline__ v8f zero8() {
  v8f z = {0.f, 0.f, 0.f, 0.f, 0.f, 0.f, 0.f, 0.f};
  return z;
}

// D(16x16,f32) = A(16x4,f32) * B(4x16,f32) + C
// A layout per lane L (m=L&15, h=L>>4): a.x=A[m][2h], a.y=A[m][2h+1]
// B layout per lane L (n=L&15, h=L>>4): b.x=B[2h][n], b.y=B[2h+1][n]
// C/D: vgpr j holds (M=j+8*(L>>4), N=L&15)
__device__ __forceinline__ v8f wmma_f32(v2f a, v2f b, v8f c) {
  return __builtin_amdgcn_wmma_f32_16x16x4_f32(false, a, false, b,
                                               (short)0, c, false, false);
}

__device__ __forceinline__ float elu1(float x) {
  return x > 0.f ? x + 1.f : __expf(x);   // elu(x)+1
}

// Normalize a [256 x kNT] slab of x into LDS (row stride kLDXN).
__device__ __forceinline__ void load_xn_tile(const float* __restrict__ x,
                                             const float* __restrict__ ws,
                                             const float* __restrict__ gamma,
                                             const float* __restrict__ beta,
                                             float* __restrict__ xn,
                                             int b, int n0, int tid) {
  constexpr int NT4 = kNT / 4;  // 16 float4 per channel row
  for (int idx = tid; idx < kC * NT4; idx += 256) {
    const int c  = idx / NT4;
    const int n4 = (idx - c * NT4) * 4;
    const int g  = c >> 5;  // c / kCPG
    const float mean = ws[WS_MRS + 2 * (b * kG + g) + 0];
    const float rstd = ws[WS_MRS + 2 * (b * kG + g) + 1];
    const float ga = gamma[c] * rstd;
    const float be = beta[c] - mean * ga;
    const float4 v = *(const float4*)(x + (size_t)(b * kC + c) * kN + n0 + n4);
    float* dst = xn + c * kLDXN + n4;
    dst[0] = v.x * ga + be;
    dst[1] = v.y * ga + be;
    dst[2] = v.z * ga + be;
    dst[3] = v.w * ga + be;
  }
}

// -------------------------- kernel 0: zero workspace -----------------------
__global__ void k_ws_init(float* __restrict__ ws) {
  const int i = blockIdx.x * 256 + threadIdx.x;
  if (i < WS_TOTAL) ws[i] = 0.f;
}

// -------------------------- kernel 1: GroupNorm stats ----------------------
// Each group is a contiguous 1M-float slab. 64 blocks per (b,g); each block
// reduces 16384 floats, then float-atomicAdd of (sum, sumsq).
__global__ void __launch_bounds__(256)
k_gn_stats(const float* __restrict__ x, float* __restrict__ ws) {
  __shared__ float red[512];
  const int bg  = blockIdx.x >> 6;   // / 64
  const int blk = blockIdx.x & 63;
  const int tid = threadIdx.x;
  const float* base = x + (size_t)bg * (kCPG * (size_t)kN) + blk * 16384;
  float s = 0.f, sq = 0.f;
#pragma unroll
  for (int i = 0; i < 16; ++i) {
    const float4 v = *(const float4*)(base + i * 1024 + tid * 4);
    s  += v.x + v.y + v.z + v.w;
    sq += v.x * v.x + v.y * v.y + v.z * v.z + v.w * v.w;
  }
  red[tid] = s;
  red[256 + tid] = sq;
  __syncthreads();
  for (int off = 128; off > 0; off >>= 1) {
    if (tid < off) {
      red[tid] += red[tid + off];
      red[256 + tid] += red[256 + tid + off];
    }
    __syncthreads();
  }
  if (tid == 0) {
    atomicAdd(&ws[WS_STAT + 2 * bg + 0], red[0]);
    atomicAdd(&ws[WS_STAT + 2 * bg + 1], red[256]);
  }
}

// -------------------------- kernel 2: finalize mean/rstd -------------------
__global__ void k_gn_finalize(float* __restrict__ ws) {
  const int i = threadIdx.x;
  if (i < kB * kG) {
    const float inv  = 1.0f / (float)(kCPG * kN);
    const float mean = ws[WS_STAT + 2 * i] * inv;
    const float var  = ws[WS_STAT + 2 * i + 1] * inv - mean * mean;
    ws[WS_MRS + 2 * i + 0] = mean;
    ws[WS_MRS + 2 * i + 1] = rsqrtf(var + 1e-5f);
  }
}

// -------------------------- kernel 3: k/v + context + k_sum ----------------
// Block = 256 threads (8 waves). Tile = 64 tokens. Wave w owns output
// channels [16w, 16w+16). Context (per head 32x32) accumulated via WMMA from
// LDS km/v tiles, flushed with float atomics at block end.
__global__ void __launch_bounds__(256)
k_attn_ctx(const float* __restrict__ x, const float* __restrict__ gamma,
           const float* __restrict__ beta, const float* __restrict__ wk,
           const float* __restrict__ wvm, float* __restrict__ ws) {
  extern __shared__ float smem[];
  float* xn  = smem;                      // kC * kLDXN
  float* kmL = xn + kC * kLDXN;           // kC * kLDT
  float* vL  = kmL + kC * kLDT;           // kC * kLDT

  const int tid  = threadIdx.x;
  const int lane = tid & 31;
  const int wid  = tid >> 5;
  const int m    = lane & 15;
  const int hh   = lane >> 4;
  const int b    = blockIdx.x >> 9;               // / kTilesPerBatch
  const int n0   = (blockIdx.x & 511) * kNT;
  const int o0   = wid * 16;

  load_xn_tile(x, ws, gamma, beta, xn, b, n0, tid);
  __syncthreads();

  v8f ctxacc[4] = {zero8(), zero8(), zero8(), zero8()};
  float ksum = 0.f;

  for (int cn = 0; cn < kNT / 16; ++cn) {
    const int nc0 = cn * 16;
    // --- k & v projection tiles: [16 o x 16 n], K-loop over 256 channels ---
    v8f kacc = zero8();
    v8f vacc = zero8();
#pragma unroll 8
    for (int c0 = 0; c0 < kC; c0 += 4) {
      const float* pk = wk  + (o0 + m) * kC + c0 + 2 * hh;
      const float* pv = wvm + (o0 + m) * kC + c0 + 2 * hh;
      v2f ak; ak.x = pk[0]; ak.y = pk[1];
      v2f av; av.x = pv[0]; av.y = pv[1];
      const float* pb = xn + (c0 + 2 * hh) * kLDXN + nc0 + m;
      v2f bx; bx.x = pb[0]; bx.y = pb[kLDXN];
      kacc = wmma_f32(ak, bx, kacc);
      vacc = wmma_f32(av, bx, vacc);
    }
    // km = elu(k)+1; stage km/v tiles in LDS for cross-wave head GEMMs
#pragma unroll
    for (int j = 0; j < 8; ++j) {
      const int row = o0 + j + 8 * hh;
      kmL[row * kLDT + m] = elu1(kacc[j]);
      vL[row * kLDT + m]  = vacc[j];
    }
    __syncthreads();

    // k_sum: thread tid owns channel tid
    {
      const float* r = kmL + tid * kLDT;
      float s = 0.f;
#pragma unroll
      for (int n = 0; n < 16; ++n) s += r[n];
      ksum += s;
    }

    // context[d][e] += sum_n km[d][n] * v[e][n]  (per head, 2x2 16x16 tiles)
    // 32 tile-jobs over 8 waves -> 4 jobs/wave, accumulators live across cn.
#pragma unroll
    for (int jj = 0; jj < 4; ++jj) {
      const int job = wid + 8 * jj;
      const int h   = job >> 2;
      const int ti  = (job >> 1) & 1;
      const int tj  = job & 1;
      const int drow = h * 32 + ti * 16;
      const int erow = h * 32 + tj * 16;
      v8f acc = ctxacc[jj];
#pragma unroll
      for (int ks = 0; ks < 4; ++ks) {
        const int kb = ks * 4;
        const float* pa = kmL + (drow + m) * kLDT + kb + 2 * hh;
        v2f a; a.x = pa[0]; a.y = pa[1];
        const float* pb = vL + (erow + m) * kLDT + kb + 2 * hh;
        v2f bb; bb.x = pb[0]; bb.y = pb[1];
        acc = wmma_f32(a, bb, acc);
      }
      ctxacc[jj] = acc;
    }
    __syncthreads();  // km/v reads done before next chunk overwrites
  }

  atomicAdd(&ws[WS_KSUM + b * kC + tid], ksum);
#pragma unroll
  for (int jj = 0; jj < 4; ++jj) {
    const int job = wid + 8 * jj;
    const int h   = job >> 2;
    const int ti  = (job >> 1) & 1;
    const int tj  = job & 1;
#pragma unroll
    for (int j = 0; j < 8; ++j) {
      const int d = ti * 16 + j + 8 * hh;
      const int e = tj * 16 + m;
      atomicAdd(&ws[WS_CTX + ((b * kH + h) * kD + d) * kD + e], ctxacc[jj][j]);
    }
  }
}

// -------------------------- kernel 4: q, numerator, out --------------------
__global__ void __launch_bounds__(256)
k_attn_out(const float* __restrict__ x, const float* __restrict__ gamma,
           const float* __restrict__ beta, const float* __restrict__ wq,
           const float* __restrict__ wout, const float* __restrict__ bout,
           const float* __restrict__ ws, float* __restrict__ out) {
  extern __shared__ float smem[];
  float* xn   = smem;                    // kC * kLDXN
  float* qmL  = xn + kC * kLDXN;         // kC * kLDT
  float* attL = qmL + kC * kLDT;         // kC * kLDT
  float* denL = attL + kC * kLDT;        // kH * 16

  const int tid  = threadIdx.x;
  const int lane = tid & 31;
  const int wid  = tid >> 5;
  const int m    = lane & 15;
  const int hh   = lane >> 4;
  const int b    = blockIdx.x >> 9;
  const int n0   = (blockIdx.x & 511) * kNT;
  const int o0   = wid * 16;
  const int h    = wid >> 1;   // head of this wave's output rows
  const int et   = wid & 1;    // e-tile within head

  const float* ctx = ws + WS_CTX + (b * kH + h) * kD * kD;
  const float* ksm = ws + WS_KSUM + b * kC;

  load_xn_tile(x, ws, gamma, beta, xn, b, n0, tid);
  __syncthreads();

  for (int cn = 0; cn < kNT / 16; ++cn) {
    const int nc0 = cn * 16;
    // --- q projection, qm = elu(q)+1 ---
    v8f qacc = zero8();
#pragma unroll 8
    for (int c0 = 0; c0 < kC; c0 += 4) {
      const float* pq = wq + (o0 + m) * kC + c0 + 2 * hh;
      v2f a; a.x = pq[0]; a.y = pq[1];
      const float* pb = xn + (c0 + 2 * hh) * kLDXN + nc0 + m;
      v2f bx; bx.x = pb[0]; bx.y = pb[kLDXN];
      qacc = wmma_f32(a, bx, qacc);
    }
#pragma unroll
    for (int j = 0; j < 8; ++j) {
      const int row = o0 + j + 8 * hh;
      qmL[row * kLDT + m] = elu1(qacc[j]);
    }
    __syncthreads();

    // denom[h][n] = sum_d qm[32h+d][n] * k_sum[32h+d]
    if (tid < kH * 16) {
      const int h2 = tid >> 4, n = tid & 15;
      float s = 0.f;
#pragma unroll 8
      for (int d = 0; d < kD; ++d)
        s += qmL[(h2 * 32 + d) * kLDT + n] * ksm[h2 * 32 + d];
      denL[tid] = s;
    }

    // numerator[e][n] = sum_d ctx[d][e] * qm[32h+d][n]  (A = ctx^T)
    v8f nacc = zero8();
#pragma unroll
    for (int d0 = 0; d0 < kD; d0 += 4) {
      const float* pa = ctx + (d0 + 2 * hh) * kD + et * 16 + m;
      v2f a; a.x = pa[0]; a.y = pa[kD];
      const float* pb = qmL + (h * 32 + d0 + 2 * hh) * kLDT + m;
      v2f bb; bb.x = pb[0]; bb.y = pb[kLDT];
      nacc = wmma_f32(a, bb, nacc);
    }
    __syncthreads();  // denom ready; qm reads complete

    const float dn = denL[(h << 4) + m] + 1e-6f;
#pragma unroll
    for (int j = 0; j < 8; ++j) {
      const int row = o0 + j + 8 * hh;
      attL[row * kLDT + m] = nacc[j] / dn;
    }
    __syncthreads();

    // out = w_out @ att + b_out + x (residual)
    v8f oacc = zero8();
#pragma unroll 8
    for (int c0 = 0; c0 < kC; c0 += 4) {
      const float* pw = wout + (o0 + m) * kC + c0 + 2 * hh;
      v2f a; a.x = pw[0]; a.y = pw[1];
      const float* pb = attL + (c0 + 2 * hh) * kLDT + m;
      v2f bb; bb.x = pb[0]; bb.y = pb[kLDT];
      oacc = wmma_f32(a, bb, oacc);
    }
#pragma unroll
    for (int j = 0; j < 8; ++j) {
      const int row = o0 + j + 8 * hh;
      const size_t gi = (size_t)(b * kC + row) * kN + n0 + nc0 + m;
      out[gi] = oacc[j] + bout[row] + x[gi];
    }
    __syncthreads();  // att/qm reads done before next chunk overwrites
  }
}

}  // namespace

extern "C" void kernel_launch(void* const* d_in, const int* in_sizes, int n_in,
                              void* d_out, int out_size, void* d_ws,
                              size_t ws_size, hipStream_t stream) {
  const float* x     = (const float*)d_in[0];
  const float* gamma = (const float*)d_in[1];
  const float* beta  = (const float*)d_in[2];
  const float* wq    = (const float*)d_in[3];
  const float* wk    = (const float*)d_in[4];
  const float* wv    = (const float*)d_in[5];
  const float* wout  = (const float*)d_in[6];
  const float* bout  = (const float*)d_in[7];
  float* out = (float*)d_out;
  float* ws  = (float*)d_ws;

  (void)in_sizes; (void)n_in; (void)out_size; (void)ws_size;

  k_ws_init<<<(WS_TOTAL + 255) / 256, 256, 0, stream>>>(ws);
  k_gn_stats<<<kB * kG * 64, 256, 0, stream>>>(x, ws);
  k_gn_finalize<<<1, 32, 0, stream>>>(ws);

  const size_t lds_b = (size_t)(kC * kLDXN + 2 * kC * kLDT) * sizeof(float);
  k_attn_ctx<<<kB * kTilesPerBatch, 256, lds_b, stream>>>(x, gamma, beta, wk,
                                                          wv, ws);
  const size_t lds_c =
      (size_t)(kC * kLDXN + 2 * kC * kLDT + kH * 16) * sizeof(float);
  k_attn_out<<<kB * kTilesPerBatch, 256, lds_c, stream>>>(x, gamma, beta, wq,
                                                          wout, bout, ws, out);
}